// GraphEncoder_81054622810762
// MI455X (gfx1250) — compile-verified
//
#include <hip/hip_runtime.h>

typedef __attribute__((ext_vector_type(2))) float v2f;
typedef __attribute__((ext_vector_type(8))) float v8f;

#define NEG_SLOPE 0.2f

// ---- order-preserving float <-> uint mapping for atomic segment-max ----
__device__ __forceinline__ unsigned fkey(float f) {
  unsigned b = __float_as_uint(f);
  return ((int)b >= 0) ? (b | 0x80000000u) : ~b;
}
__device__ __forceinline__ float funkey(unsigned k) {
  return (k & 0x80000000u) ? __uint_as_float(k & 0x7FFFFFFFu) : __uint_as_float(~k);
}

// ---------------------------------------------------------------------------
// Generic f32 GEMM on the matrix pipe: out[M,Nout] = act(A[M,K] @ W[K,Nout] + b)
// One wave computes one 16x16 output tile via V_WMMA_F32_16X16X4_F32.
// A-frag layout: lanes 0-15 -> (row=lane, K=k0/k0+1); lanes 16-31 -> K=k0+2/k0+3.
// C/D layout: VGPR r -> rows r (lanes 0-15) and r+8 (lanes 16-31).
// M, Nout must be multiples of 16; grids sized so guard is wave-uniform.
// ---------------------------------------------------------------------------
__global__ void gemm16_wmma_f32(const float* __restrict__ A, const float* __restrict__ W,
                                const float* __restrict__ bias, float* __restrict__ out,
                                int M, int K, int Nout, int relu) {
  int wave = (int)(((long)blockIdx.x * blockDim.x + threadIdx.x) >> 5);
  int lane = (int)(threadIdx.x & 31);
  int ntiles = Nout >> 4;
  int tiles  = (M >> 4) * ntiles;
  if (wave >= tiles) return;                 // wave-uniform: EXEC stays all-ones
  int mt = wave / ntiles;
  int nt = wave - mt * ntiles;
  int row = (mt << 4) + (lane & 15);
  int kh  = (lane >> 4) << 1;                // 0 (lanes 0-15) or 2 (lanes 16-31)
  int col = (nt << 4) + (lane & 15);
  const float* Ar = A + (size_t)row * K;
  v8f acc = {};
  for (int k0 = 0; k0 < K; k0 += 4) {
    v2f a, b;
    a.x = Ar[k0 + kh];
    a.y = Ar[k0 + kh + 1];
    b.x = W[(size_t)(k0 + kh) * Nout + col];
    b.y = W[(size_t)(k0 + kh + 1) * Nout + col];
    acc = __builtin_amdgcn_wmma_f32_16x16x4_f32(false, a, false, b, (short)0, acc,
                                                false, false);
  }
  float bv = bias ? bias[col] : 0.f;
  int rbase = (mt << 4) + ((lane >> 4) << 3);
#pragma unroll
  for (int r = 0; r < 8; ++r) {
    float v = acc[r] + bv;
    if (relu) v = fmaxf(v, 0.f);
    out[(size_t)(rbase + r) * Nout + col] = v;
  }
}

// ---- self-loop attr: sum incoming edge_attr per dst + in-degree -----------
__global__ void loop_accum_k(const int* __restrict__ ei, const float* __restrict__ ea,
                             float* __restrict__ lsum, float* __restrict__ cnt, int E) {
  long t = (long)blockIdx.x * blockDim.x + threadIdx.x;
  if (t >= (long)E * 16) return;
  int e = (int)(t >> 4), f = (int)(t & 15);
  int dst = ei[E + e];
  atomicAdd(&lsum[(size_t)dst * 16 + f], ea[t]);
  if (f == 0) atomicAdd(&cnt[dst], 1.f);
}

__global__ void loop_div_k(float* __restrict__ lsum, const float* __restrict__ cnt, int N) {
  long t = (long)blockIdx.x * blockDim.x + threadIdx.x;
  if (t >= (long)N * 16) return;
  lsum[t] /= fmaxf(cnt[t >> 4], 1.f);
}

// ---- pass 1: per-edge attention logits + scatter-max ----------------------
// One wave per edge; lane handles channels c and c+32 (H=4, C=16 -> 64 chans).
__global__ void edge_score_k(const float* __restrict__ xl, const float* __restrict__ xr,
                             const int* __restrict__ ei, const float* __restrict__ ea,
                             const float* __restrict__ lattr, const float* __restrict__ We,
                             const float* __restrict__ att, float* __restrict__ score,
                             unsigned* __restrict__ smax, int E, int N) {
  long gid = (long)blockIdx.x * blockDim.x + threadIdx.x;
  int wid = (int)(gid >> 5);
  int lane = (int)(threadIdx.x & 31);
  int E2 = E + N;
  if (wid >= E2) return;
  int src, dst;
  const float* eav;
  if (wid < E) { src = ei[wid]; dst = ei[E + wid]; eav = ea + (size_t)wid * 16; }
  else         { src = wid - E; dst = src;         eav = lattr + (size_t)src * 16; }
  int c0 = lane, c1 = lane + 32;
  float e0 = 0.f, e1 = 0.f;
#pragma unroll
  for (int k = 0; k < 16; ++k) {
    float a = eav[k];                       // uniform across lanes (cache broadcast)
    e0 = fmaf(a, We[k * 64 + c0], e0);
    e1 = fmaf(a, We[k * 64 + c1], e1);
  }
  float m0 = xl[(size_t)src * 64 + c0] + xr[(size_t)dst * 64 + c0] + e0;
  float m1 = xl[(size_t)src * 64 + c1] + xr[(size_t)dst * 64 + c1] + e1;
  m0 = (m0 > 0.f) ? m0 : m0 * NEG_SLOPE;
  m1 = (m1 > 0.f) ? m1 : m1 * NEG_SLOPE;
  float s0 = m0 * att[c0];                  // att flat [H*C]=64, index == channel
  float s1 = m1 * att[c1];
#pragma unroll
  for (int off = 8; off >= 1; off >>= 1) {  // reduce within each 16-lane head group
    s0 += __shfl_xor(s0, off, 32);
    s1 += __shfl_xor(s1, off, 32);
  }
  if ((lane & 15) == 0) {
    int h0 = lane >> 4;                     // head 0 or 1 (c1 part is head h0+2)
    score[(size_t)wid * 4 + h0] = s0;
    score[(size_t)wid * 4 + h0 + 2] = s1;
    atomicMax(&smax[(size_t)dst * 4 + h0], fkey(s0));
    atomicMax(&smax[(size_t)dst * 4 + h0 + 2], fkey(s1));
  }
}

// ---- pass 2: z = exp(score - max), scatter-sum denom; z cached in-place ----
__global__ void edge_z_k(const int* __restrict__ ei, float* __restrict__ zsc,
                         const unsigned* __restrict__ smax, float* __restrict__ denom,
                         int E, int N) {
  long t = (long)blockIdx.x * blockDim.x + threadIdx.x;
  if (t >= (long)(E + N) * 4) return;
  int e = (int)(t >> 2), h = (int)(t & 3);
  int dst = (e < E) ? ei[E + e] : (e - E);
  float mx = funkey(smax[(size_t)dst * 4 + h]);
  float z = __expf(zsc[t] - mx);
  zsc[t] = z;
  atomicAdd(&denom[(size_t)dst * 4 + h], z);
}

// ---- pass 3: out[dst] += alpha * xl[src] (wave per edge) ------------------
__global__ void edge_aggr_k(const float* __restrict__ xl, const int* __restrict__ ei,
                            const float* __restrict__ zsc, const float* __restrict__ denom,
                            float* __restrict__ accum, int E, int N) {
  long gid = (long)blockIdx.x * blockDim.x + threadIdx.x;
  int wid = (int)(gid >> 5);
  int lane = (int)(threadIdx.x & 31);
  int E2 = E + N;
  if (wid >= E2) return;
  int src, dst;
  if (wid < E) { src = ei[wid]; dst = ei[E + wid]; }
  else         { src = wid - E; dst = src; }
  int c0 = lane, c1 = lane + 32;
  float a0 = zsc[(size_t)wid * 4 + (c0 >> 4)] / denom[(size_t)dst * 4 + (c0 >> 4)];
  float a1 = zsc[(size_t)wid * 4 + (c1 >> 4)] / denom[(size_t)dst * 4 + (c1 >> 4)];
  atomicAdd(&accum[(size_t)dst * 64 + c0], a0 * xl[(size_t)src * 64 + c0]);
  atomicAdd(&accum[(size_t)dst * 64 + c1], a1 * xl[(size_t)src * 64 + c1]);
}

// ---- per-node finalize: relu(accum + bias); optional mean-pool scatter -----
__global__ void finalize_k(const float* __restrict__ accum, const float* __restrict__ bias,
                           float* __restrict__ xout, const int* __restrict__ batch,
                           float* __restrict__ gsum, float* __restrict__ gcnt,
                           int N, int pool) {
  long t = (long)blockIdx.x * blockDim.x + threadIdx.x;
  if (t >= (long)N * 64) return;
  int n = (int)(t >> 6), c = (int)(t & 63);
  float v = fmaxf(accum[t] + bias[c], 0.f);
  xout[t] = v;
  if (pool) {
    int g = batch[n];
    atomicAdd(&gsum[(size_t)g * 64 + c], v);
    if (c == 0) atomicAdd(&gcnt[g], 1.f);
  }
}

__global__ void gdiv_k(float* __restrict__ gsum, const float* __restrict__ gcnt, int G) {
  int t = blockIdx.x * blockDim.x + threadIdx.x;
  if (t >= G * 64) return;
  gsum[t] /= fmaxf(gcnt[t >> 6], 1.f);
}

// ---------------------------------------------------------------------------
extern "C" void kernel_launch(void* const* d_in, const int* in_sizes, int n_in,
                              void* d_out, int out_size, void* d_ws, size_t ws_size,
                              hipStream_t stream) {
  const float* x0    = (const float*)d_in[0];
  const int*   ei    = (const int*)d_in[1];
  const float* ea    = (const float*)d_in[2];
  const int*   batch = (const int*)d_in[3];
  int N  = in_sizes[0] / 64;
  int E  = in_sizes[1] / 2;
  int G  = out_size / 128;
  int E2 = E + N;

  float* ws = (float*)d_ws;
  size_t o = 0;
  float*    cnt   = ws + o; o += (size_t)N;          // in-degree (float)
  float*    lattr = ws + o; o += (size_t)N * 16;     // self-loop attrs
  float*    xl    = ws + o; o += (size_t)N * 64;
  float*    xr    = ws + o; o += (size_t)N * 64;
  float*    x1    = ws + o; o += (size_t)N * 64;     // layer output
  unsigned* smax  = (unsigned*)(ws + o); o += (size_t)N * 4;   // } contiguous
  float*    denom = ws + o; o += (size_t)N * 4;                // } zero block
  float*    accum = ws + o; o += (size_t)N * 64;               // } (N*72 floats)
  float*    zsc   = ws + o; o += (size_t)E2 * 4;
  float*    gsum  = ws + o; o += (size_t)G * 64;     // } contiguous (G*65)
  float*    gcnt  = ws + o; o += (size_t)G;          // }
  float*    h1    = ws + o; o += (size_t)G * 128;

  const int B = 256;

  // self-loop attr = mean of incoming edge attrs
  hipMemsetAsync(cnt,  0, sizeof(float) * (size_t)N * 17, stream);   // cnt+lattr
  hipMemsetAsync(gsum, 0, sizeof(float) * (size_t)G * 65, stream);   // gsum+gcnt
  {
    long t = (long)E * 16;
    loop_accum_k<<<(int)((t + B - 1) / B), B, 0, stream>>>(ei, ea, lattr, cnt, E);
    long t2 = (long)N * 16;
    loop_div_k<<<(int)((t2 + B - 1) / B), B, 0, stream>>>(lattr, cnt, N);
  }

  for (int l = 0; l < 2; ++l) {
    const float* xin  = l ? x1 : x0;
    int base = 4 + l * 7;
    const float* Wl   = (const float*)d_in[base + 0];
    const float* bl   = (const float*)d_in[base + 1];
    const float* Wr   = (const float*)d_in[base + 2];
    const float* br   = (const float*)d_in[base + 3];
    const float* We   = (const float*)d_in[base + 4];
    const float* att  = (const float*)d_in[base + 5];
    const float* bias = (const float*)d_in[base + 6];

    hipMemsetAsync(smax, 0, sizeof(float) * (size_t)N * 72, stream); // smax+denom+accum

    // node transforms on the matrix pipe (f32 WMMA, exact precision)
    int tiles = (N / 16) * (64 / 16);
    int grid  = (tiles * 32 + B - 1) / B;
    gemm16_wmma_f32<<<grid, B, 0, stream>>>(xin, Wl, bl, xl, N, 64, 64, 0);
    gemm16_wmma_f32<<<grid, B, 0, stream>>>(xin, Wr, br, xr, N, 64, 64, 0);

    long ethreads = (long)E2 * 32;
    int  egrid    = (int)((ethreads + B - 1) / B);
    edge_score_k<<<egrid, B, 0, stream>>>(xl, xr, ei, ea, lattr, We, att, zsc, smax, E, N);
    long zthreads = (long)E2 * 4;
    edge_z_k<<<(int)((zthreads + B - 1) / B), B, 0, stream>>>(ei, zsc, smax, denom, E, N);
    edge_aggr_k<<<egrid, B, 0, stream>>>(xl, ei, zsc, denom, accum, E, N);
    long fthreads = (long)N * 64;
    finalize_k<<<(int)((fthreads + B - 1) / B), B, 0, stream>>>(accum, bias, x1, batch,
                                                               gsum, gcnt, N, l == 1);
  }

  // global mean pool + MLP head (WMMA)
  gdiv_k<<<(G * 64 + B - 1) / B, B, 0, stream>>>(gsum, gcnt, G);

  const float* W1 = (const float*)d_in[18];
  const float* b1 = (const float*)d_in[19];
  const float* W2 = (const float*)d_in[20];
  const float* b2 = (const float*)d_in[21];
  float* outp = (float*)d_out;
  {
    int tiles = (G / 16) * (128 / 16);
    int grid  = (tiles * 32 + B - 1) / B;
    gemm16_wmma_f32<<<grid, B, 0, stream>>>(gsum, W1, b1, h1,   G, 64,  128, 1);
    gemm16_wmma_f32<<<grid, B, 0, stream>>>(h1,   W2, b2, outp, G, 128, 128, 0);
  }
}